// RelativePositionEncoding_7619271983169
// MI455X (gfx1250) — compile-verified
//
#include <hip/hip_runtime.h>

// Native vector types (support arithmetic + __builtin_nontemporal_store)
typedef float f4 __attribute__((ext_vector_type(4)));
typedef int   i4 __attribute__((ext_vector_type(4)));

#define N_TOK  1024
#define C_PAIR 128
#define R_MAX  32
#define S_MAX  2
#define N_R    (2 * R_MAX + 2)          // 66
#define NF     (2 * N_R + 6 + 1)        // 139 feature rows in W^T

// ---- CDNA5 async global->LDS copy (per-lane LDS dst addr, per-lane global src) ----
__device__ __forceinline__ unsigned lds_addr(const void* p) {
  // generic pointer to LDS: low 32 bits are the workgroup-relative LDS byte offset
  return (unsigned)(unsigned long long)p;
}
__device__ __forceinline__ void async_copy_b32(unsigned lds, const void* g) {
  asm volatile("global_load_async_to_lds_b32 %0, %1, off" :: "v"(lds), "v"(g) : "memory");
}
__device__ __forceinline__ void async_copy_b128(unsigned lds, const void* g) {
  asm volatile("global_load_async_to_lds_b128 %0, %1, off" :: "v"(lds), "v"(g) : "memory");
}
__device__ __forceinline__ void wait_async0() {
  asm volatile("s_wait_asynccnt 0" ::: "memory");
}

__global__ __launch_bounds__(256)
void relpos_enc_kernel(const int* __restrict__ residue,
                       const int* __restrict__ token,
                       const int* __restrict__ asym,
                       const int* __restrict__ entity,
                       const float* __restrict__ maskp,
                       const float* __restrict__ W,     // (128, 139) row-major
                       float* __restrict__ out,         // (N, N, 128)
                       int N)
{
  __shared__ __align__(16) float sW[NF * C_PAIR];    // W^T: [f][c], 71168 B
  __shared__ __align__(16) float sCE[12 * C_PAIR];   // fused W_chain + e*W_ent: [e*6+s][c]
  __shared__ __align__(16) i4    sPack[N_TOK];       // {residue, token, asym, entity}
  __shared__ __align__(16) float sMask[N_TOK];

  const int tid = threadIdx.x;
  const int i   = blockIdx.x;

  // ---- Stage W transposed into LDS via CDNA5 async-tensor path ----
  // global element n = c*NF + f  ->  LDS slot f*C_PAIR + c  (transpose done by the DMA)
  for (int n = tid; n < C_PAIR * NF; n += 256) {
    int c = n / NF;
    int f = n - c * NF;
    async_copy_b32(lds_addr(&sW[f * C_PAIR + c]), W + n);
  }
  // ---- Stage mask (contiguous, 128-bit async copies) ----
  for (int n = tid; n < N_TOK / 4; n += 256) {
    async_copy_b128(lds_addr(&sMask[n * 4]), maskp + n * 4);
  }
  // ---- Pack per-token integer features (plain coalesced loads + ds stores) ----
  for (int j = tid; j < N_TOK; j += 256) {
    i4 v;
    v.x = residue[j]; v.y = token[j]; v.z = asym[j]; v.w = entity[j];
    sPack[j] = v;
  }

  // Per-row (uniform) scalars for this block's i
  const int   ri = residue[i], ti = token[i], ai = asym[i], ei = entity[i];
  const float mi = maskp[i];

  wait_async0();       // wave's async copies have landed in LDS
  __syncthreads();     // all waves' copies visible

  // ---- Build fused chain/entity table: sCE[e*6+s] = W_chain[s] + e*W_ent ----
  for (int n = tid; n < 12 * C_PAIR; n += 256) {
    int row = n >> 7;            // 0..11
    int c   = n & (C_PAIR - 1);
    int e   = row >= 6;
    int s   = row - (e ? 6 : 0);
    float v = sW[(2 * N_R + 1 + s) * C_PAIR + c];
    if (e) v += sW[(2 * N_R) * C_PAIR + c];
    sCE[n] = v;
  }
  __syncthreads();

  // ---- Main loop: one wave per (i,j) pair; lane l covers channels 4l..4l+3 ----
  const int wave = tid >> 5;
  const int lane = tid & 31;
  float* outRow = out + (size_t)i * N * C_PAIR;

#pragma unroll 2
  for (int j = wave; j < N; j += 8) {
    i4    s  = sPack[j];     // LDS broadcast (all lanes same address)
    float mj = sMask[j];

    bool sch  = (ai == s.z);
    bool sres = (ri == s.x);
    bool sent = (ei == s.w);

    // d[i,j] = feat[j] - feat[i]
    int dp = sch          ? min(max(s.x - ri + R_MAX, 0), 2 * R_MAX) : 2 * R_MAX + 1;
    int dt = (sch && sres)? min(max(s.y - ti + R_MAX, 0), 2 * R_MAX) : 2 * R_MAX + 1;
    int dc = sch          ? min(max(s.z - ai + S_MAX, 0), 2 * S_MAX) : 2 * S_MAX + 1;
    int ce = dc + (sent ? 6 : 0);
    float m = mi * mj;

    f4 a = ((const f4*)(sW  + dp         * C_PAIR))[lane];   // W_pos[d_pos]
    f4 b = ((const f4*)(sW  + (N_R + dt) * C_PAIR))[lane];   // W_tok[d_tok]
    f4 c = ((const f4*)(sCE + ce         * C_PAIR))[lane];   // W_chain[d_chain] + e*W_ent

    f4 o = (a + b + c) * m;

    // streaming 128-bit store, non-temporal (output >> L2)
    __builtin_nontemporal_store(o, (f4*)(outRow + (size_t)j * C_PAIR) + lane);
  }
}

extern "C" void kernel_launch(void* const* d_in, const int* in_sizes, int n_in,
                              void* d_out, int out_size, void* d_ws, size_t ws_size,
                              hipStream_t stream) {
  const int*   residue = (const int*)d_in[0];
  const int*   token   = (const int*)d_in[1];
  const int*   asym    = (const int*)d_in[2];
  const int*   entity  = (const int*)d_in[3];
  const float* mask    = (const float*)d_in[4];
  const float* W       = (const float*)d_in[5];
  float*       out     = (float*)d_out;

  const int N = in_sizes[0];   // 1024 (B==1 in reference)
  relpos_enc_kernel<<<N, 256, 0, stream>>>(residue, token, asym, entity, mask, W, out, N);
}